// MoEGate_4930622456424
// MI455X (gfx1250) — compile-verified
//
#include <hip/hip_runtime.h>

typedef float v2f __attribute__((ext_vector_type(2)));
typedef float v8f __attribute__((ext_vector_type(8)));

#define HDIM 4096
#define NEXP 64
#define TOPK 8
#define TOK_PER_BLOCK 128
#define LDS_STRIDE 65   // 64 + 1 pad -> conflict-free per-token row reads
#define ALPHA 0.001f

// ---------------------------------------------------------------------------
// Kernel 0: zero the workspace accumulators (per-expert score sums + counts)
// ---------------------------------------------------------------------------
__global__ void moe_zero_ws(float* __restrict__ ws) {
    int i = threadIdx.x;
    if (i < 2 * NEXP) ws[i] = 0.0f;
}

// ---------------------------------------------------------------------------
// Kernel 1: router GEMM (WMMA f32 16x16x4) + softmax + top-8 + aux partials
// ---------------------------------------------------------------------------
__global__ __launch_bounds__(256)
void moe_gate_main(const float* __restrict__ x,      // [T, HDIM]
                   const float* __restrict__ w,      // [NEXP, HDIM]
                   float* __restrict__ out_w,        // [T, TOPK]
                   float* __restrict__ out_idx,      // [T, TOPK] (as float)
                   float* __restrict__ ws_sum,       // [NEXP]
                   float* __restrict__ ws_cnt) {     // [NEXP]
    __shared__ float lds_logits[TOK_PER_BLOCK * LDS_STRIDE];
    __shared__ float lds_sum[NEXP];
    __shared__ float lds_cnt[NEXP];

    const int tid  = threadIdx.x;
    const int wave = tid >> 5;
    const int lane = tid & 31;

    if (tid < NEXP) { lds_sum[tid] = 0.0f; lds_cnt[tid] = 0.0f; }

    // ---- WMMA GEMM: this wave -> 16 tokens x 64 experts -------------------
    // A 16x4 f32 layout: lane holds M = lane&15; VGPR pair = K = (lane>>4)*2, +1
    // B 4x16 f32 layout: lane holds N = lane&15; VGPR pair = K = (lane>>4)*2, +1
    const int tok_base = blockIdx.x * TOK_PER_BLOCK + wave * 16;
    const int mn   = lane & 15;
    const int koff = (lane >> 4) * 2;

    const float* aptr = x + (size_t)(tok_base + mn) * HDIM + koff;
    const float* bptr0 = w + (size_t)(mn +  0) * HDIM + koff;
    const float* bptr1 = w + (size_t)(mn + 16) * HDIM + koff;
    const float* bptr2 = w + (size_t)(mn + 32) * HDIM + koff;
    const float* bptr3 = w + (size_t)(mn + 48) * HDIM + koff;

    v8f acc0 = {}, acc1 = {}, acc2 = {}, acc3 = {};

#pragma unroll 4
    for (int k = 0; k < HDIM; k += 4) {
        v2f a  = *(const v2f*)(aptr  + k);
        v2f b0 = *(const v2f*)(bptr0 + k);
        v2f b1 = *(const v2f*)(bptr1 + k);
        v2f b2 = *(const v2f*)(bptr2 + k);
        v2f b3 = *(const v2f*)(bptr3 + k);
        acc0 = __builtin_amdgcn_wmma_f32_16x16x4_f32(false, a, false, b0, (short)0, acc0, false, false);
        acc1 = __builtin_amdgcn_wmma_f32_16x16x4_f32(false, a, false, b1, (short)0, acc1, false, false);
        acc2 = __builtin_amdgcn_wmma_f32_16x16x4_f32(false, a, false, b2, (short)0, acc2, false, false);
        acc3 = __builtin_amdgcn_wmma_f32_16x16x4_f32(false, a, false, b3, (short)0, acc3, false, false);
    }

    // ---- scatter C tiles to LDS ------------------------------------------
    // C/D layout: lane holds N = lane&15; VGPR j holds M = j + (lane>>4)*8
    {
        const int row0 = wave * 16 + (lane >> 4) * 8;
        const int col  = lane & 15;
#pragma unroll
        for (int j = 0; j < 8; ++j) {
            float* r = &lds_logits[(row0 + j) * LDS_STRIDE];
            r[col +  0] = acc0[j];
            r[col + 16] = acc1[j];
            r[col + 32] = acc2[j];
            r[col + 48] = acc3[j];
        }
    }
    __syncthreads();

    // ---- per-token softmax + top-8 ---------------------------------------
    if (tid < TOK_PER_BLOCK) {
        const size_t t = (size_t)blockIdx.x * TOK_PER_BLOCK + tid;
        float s[NEXP];
        float mx = -3.0e38f;
#pragma unroll
        for (int e = 0; e < NEXP; ++e) {
            s[e] = lds_logits[tid * LDS_STRIDE + e];
            mx = fmaxf(mx, s[e]);
        }
        float sum = 0.0f;
#pragma unroll
        for (int e = 0; e < NEXP; ++e) {
            s[e] = __expf(s[e] - mx);
            sum += s[e];
        }
        const float inv = 1.0f / sum;
#pragma unroll
        for (int e = 0; e < NEXP; ++e) {
            s[e] *= inv;
            atomicAdd(&lds_sum[e], s[e]);   // Pi numerator partial
        }
        // iterative argmax -> sorted-descending top-8 (matches lax.top_k)
        unsigned long long sel = 0ull;
        for (int kk = 0; kk < TOPK; ++kk) {
            float best = -1.0f;
            int   bi   = 0;
#pragma unroll
            for (int e = 0; e < NEXP; ++e) {
                bool taken  = (sel >> e) & 1ull;
                bool better = (!taken) && (s[e] > best);
                best = better ? s[e] : best;
                bi   = better ? e    : bi;
            }
            sel |= (1ull << bi);
            out_w  [t * TOPK + kk] = best;
            out_idx[t * TOPK + kk] = (float)bi;
            atomicAdd(&lds_cnt[bi], 1.0f);
        }
    }
    __syncthreads();

    // ---- flush block partials to global workspace ------------------------
    if (tid < NEXP) {
        atomicAdd(&ws_sum[tid], lds_sum[tid]);
        atomicAdd(&ws_cnt[tid], lds_cnt[tid]);
    }
}

// ---------------------------------------------------------------------------
// Kernel 2: finalize aux loss = ALPHA * sum_e Pi[e] * fi[e]
// ---------------------------------------------------------------------------
__global__ void moe_finalize(const float* __restrict__ ws_sum,
                             const float* __restrict__ ws_cnt,
                             float* __restrict__ out_aux, int T) {
    __shared__ float terms[NEXP];
    int e = threadIdx.x;
    if (e < NEXP) {
        float Pi = ws_sum[e] / (float)T;
        float fi = ws_cnt[e] * (float)NEXP / ((float)T * (float)TOPK);
        terms[e] = Pi * fi;
    }
    __syncthreads();
    if (e == 0) {
        float total = 0.0f;
        for (int i = 0; i < NEXP; ++i) total += terms[i];
        *out_aux = total * ALPHA;
    }
}

// ---------------------------------------------------------------------------
extern "C" void kernel_launch(void* const* d_in, const int* in_sizes, int n_in,
                              void* d_out, int out_size, void* d_ws, size_t ws_size,
                              hipStream_t stream) {
    const float* x = (const float*)d_in[0];   // hidden_states [4,4096,4096]
    const float* w = (const float*)d_in[1];   // weight        [64,4096]
    const int T = in_sizes[0] / HDIM;         // 16384 tokens

    float* out     = (float*)d_out;
    float* out_w   = out;                           // [T, 8]
    float* out_idx = out + (size_t)T * TOPK;        // [T, 8]
    float* out_aux = out + (size_t)2 * T * TOPK;    // scalar

    float* ws_sum = (float*)d_ws;           // [64]
    float* ws_cnt = ws_sum + NEXP;          // [64]

    moe_zero_ws<<<1, 128, 0, stream>>>(ws_sum);
    moe_gate_main<<<T / TOK_PER_BLOCK, 256, 0, stream>>>(x, w, out_w, out_idx,
                                                         ws_sum, ws_cnt);
    moe_finalize<<<1, NEXP, 0, stream>>>(ws_sum, ws_cnt, out_aux, T);
}